// GATLayer_25632364822804
// MI455X (gfx1250) — compile-verified
//
#include <hip/hip_runtime.h>

typedef __attribute__((ext_vector_type(2))) float v2f;
typedef __attribute__((ext_vector_type(8))) float v8f;

#define N_NODES 50000
#define N_EDGES 800000
#define IN_DIM  128
#define HEADS   8
#define HEAD_DIM 16
#define CH      128            // HEADS*HEAD_DIM
#define NEG_SLOPE 0.2f
#define BN_EPS  1e-5f

// ---- workspace layout (float offsets) -------------------------------------
#define OFF_FEAT 0ull               // 50000*128      = 6,400,000
#define OFF_EL   6400000ull         // 50000*8        =   400,000
#define OFF_ER   6800000ull         //                =   400,000
#define OFF_E    7200000ull         // 800000*8       = 6,400,000 (logits->ee)
#define OFF_EMAX 13600000ull        //                =   400,000
#define OFF_DEN  14000000ull        //                =   400,000
#define OFF_AGG  14400000ull        //                = 6,400,000
#define OFF_MEAN 20800000ull        // 128
#define OFF_RSTD 20800128ull        // 128
// total 20,800,256 floats ~ 83.2 MB

// ---------------------------------------------------------------------------
// 1) feat = h @ W  via V_WMMA_F32_16X16X4_F32. One 16x16 output tile / wave.
//    50000/16 = 3125 M-tiles, 128/16 = 8 N-tiles -> 25000 waves = 3125 blocks.
// ---------------------------------------------------------------------------
__global__ __launch_bounds__(256) void gemm_feat_wmma(
    const float* __restrict__ h, const float* __restrict__ W,
    float* __restrict__ feat) {
  const int lane   = threadIdx.x & 31;
  const int gwave  = blockIdx.x * 8 + (threadIdx.x >> 5);
  const int tm     = gwave >> 3;      // 0..3124
  const int tn     = gwave & 7;       // 0..7
  const int half   = lane >> 4;       // 0 | 1
  const int l15    = lane & 15;
  const int m      = tm * 16 + l15;   // A row for this lane
  const int n      = tn * 16 + l15;   // B col for this lane
  const int kk     = half * 2;        // lanes 16-31 carry K+2,K+3

  const float* __restrict__ hrow = h + (size_t)m * IN_DIM;
  v8f acc = {};
#pragma unroll
  for (int k = 0; k < IN_DIM; k += 4) {
    v2f a, b;
    // A 16x4 f32 layout: lanes0-15 VGPR0=K0,VGPR1=K1; lanes16-31 K2,K3
    a.x = hrow[k + kk];
    a.y = hrow[k + kk + 1];
    // B 4x16 f32 layout (mirror): row K striped across lanes within a VGPR
    b.x = W[(size_t)(k + kk)     * CH + n];
    b.y = W[(size_t)(k + kk + 1) * CH + n];
    acc = __builtin_amdgcn_wmma_f32_16x16x4_f32(
        /*neg_a=*/false, a, /*neg_b=*/false, b,
        /*c_mod=*/(short)0, acc, /*reuse_a=*/false, /*reuse_b=*/false);
  }
  // C/D layout: VGPR v -> M = v + 8*half ; N = l15
  const int row0 = tm * 16 + half * 8;
  const int col  = tn * 16 + l15;
#pragma unroll
  for (int v = 0; v < 8; ++v)
    feat[(size_t)(row0 + v) * CH + col] = acc[v];
}

// ---------------------------------------------------------------------------
// 2) el/er per (node, head)
// ---------------------------------------------------------------------------
__global__ void elr_kernel(const float* __restrict__ feat,
                           const float* __restrict__ attn_l,
                           const float* __restrict__ attn_r,
                           float* __restrict__ el, float* __restrict__ er) {
  const int gid = blockIdx.x * blockDim.x + threadIdx.x;
  if (gid >= N_NODES * HEADS) return;
  const int node = gid >> 3, hd = gid & 7;
  const float* f  = feat + (size_t)node * CH + hd * HEAD_DIM;
  const float* al = attn_l + hd * HEAD_DIM;
  const float* ar = attn_r + hd * HEAD_DIM;
  float sl = 0.f, sr = 0.f;
#pragma unroll
  for (int d = 0; d < HEAD_DIM; ++d) { sl += f[d] * al[d]; sr += f[d] * ar[d]; }
  el[gid] = sl; er[gid] = sr;
}

// ---------------------------------------------------------------------------
// 3) init scratch: emax=-inf, denom=0, agg=0
// ---------------------------------------------------------------------------
__global__ void init_kernel(float* __restrict__ emax, float* __restrict__ den,
                            float* __restrict__ agg) {
  const int gid = blockIdx.x * blockDim.x + threadIdx.x;
  if (gid < N_NODES * HEADS) { emax[gid] = -__builtin_inff(); den[gid] = 0.f; }
  if (gid < N_NODES * CH)    agg[gid] = 0.f;
}

// float atomic max via int/uint ordering trick (valid: init=-inf, no NaNs)
__device__ __forceinline__ void atomicMaxF(float* addr, float v) {
  if (v >= 0.f) atomicMax((int*)addr, __float_as_int(v));
  else          atomicMin((unsigned int*)addr, __float_as_uint(v));
}

// ---------------------------------------------------------------------------
// 4) per-(edge,head) logits + segment max
// ---------------------------------------------------------------------------
__global__ void edge_logits(const int* __restrict__ src, const int* __restrict__ dst,
                            const float* __restrict__ el, const float* __restrict__ er,
                            float* __restrict__ ebuf, float* __restrict__ emax) {
  const int gid = blockIdx.x * blockDim.x + threadIdx.x;
  if (gid >= N_EDGES * HEADS) return;
  const int e = gid >> 3, hd = gid & 7;
  const int s = src[e], d0 = dst[e];
  float x = el[s * HEADS + hd] + er[d0 * HEADS + hd];
  x = x > 0.f ? x : NEG_SLOPE * x;            // LeakyReLU
  ebuf[gid] = x;
  atomicMaxF(&emax[d0 * HEADS + hd], x);
}

// ---------------------------------------------------------------------------
// 5) ee = exp(e - emax[dst]); segment sum
// ---------------------------------------------------------------------------
__global__ void edge_exp(const int* __restrict__ dst, float* __restrict__ ebuf,
                         const float* __restrict__ emax, float* __restrict__ den) {
  const int gid = blockIdx.x * blockDim.x + threadIdx.x;
  if (gid >= N_EDGES * HEADS) return;
  const int e = gid >> 3, hd = gid & 7;
  const int d0 = dst[e];
  const float ee = __expf(ebuf[gid] - emax[d0 * HEADS + hd]);
  ebuf[gid] = ee;
  atomicAdd(&den[d0 * HEADS + hd], ee);
}

// ---------------------------------------------------------------------------
// 6) agg[dst] += alpha * feat[src]   (thread per (edge, channel))
// ---------------------------------------------------------------------------
__global__ void edge_aggregate(const int* __restrict__ src, const int* __restrict__ dst,
                               const float* __restrict__ feat,
                               const float* __restrict__ ebuf,
                               const float* __restrict__ den,
                               float* __restrict__ agg) {
  const long long gid = (long long)blockIdx.x * blockDim.x + threadIdx.x;
  if (gid >= (long long)N_EDGES * CH) return;
  const int e  = (int)(gid >> 7);
  const int c  = (int)(gid & 127);
  const int hd = c >> 4;
  const int s = src[e], d0 = dst[e];
  const float alpha = ebuf[(size_t)e * HEADS + hd] / den[d0 * HEADS + hd];
  atomicAdd(&agg[(size_t)d0 * CH + c], alpha * feat[(size_t)s * CH + c]);
}

// ---------------------------------------------------------------------------
// 7) BatchNorm statistics: one block per channel
// ---------------------------------------------------------------------------
__global__ __launch_bounds__(256) void bn_stats(const float* __restrict__ agg,
                                                float* __restrict__ mean,
                                                float* __restrict__ rstd) {
  __shared__ float ss[256], sq[256];
  const int c = blockIdx.x, t = threadIdx.x;
  float s = 0.f, q = 0.f;
  for (int i = t; i < N_NODES; i += 256) {
    const float v = agg[(size_t)i * CH + c];
    s += v; q += v * v;
  }
  ss[t] = s; sq[t] = q;
  __syncthreads();
  for (int o = 128; o > 0; o >>= 1) {
    if (t < o) { ss[t] += ss[t + o]; sq[t] += sq[t + o]; }
    __syncthreads();
  }
  if (t == 0) {
    const float m = ss[0] / (float)N_NODES;
    const float var = sq[0] / (float)N_NODES - m * m;   // biased, matches jnp.var
    mean[c] = m;
    rstd[c] = rsqrtf(var + BN_EPS);
  }
}

// ---------------------------------------------------------------------------
// 8) out = h + elu(gamma*(agg-mean)*rstd + beta)
// ---------------------------------------------------------------------------
__global__ void finalize(const float* __restrict__ h, const float* __restrict__ agg,
                         const float* __restrict__ mean, const float* __restrict__ rstd,
                         const float* __restrict__ gamma, const float* __restrict__ beta,
                         float* __restrict__ out) {
  const int gid = blockIdx.x * blockDim.x + threadIdx.x;
  if (gid >= N_NODES * CH) return;
  const int c = gid & 127;
  const float x = gamma[c] * (agg[gid] - mean[c]) * rstd[c] + beta[c];
  const float y = x > 0.f ? x : (__expf(x) - 1.f);
  out[gid] = h[gid] + y;
}

// ---------------------------------------------------------------------------
extern "C" void kernel_launch(void* const* d_in, const int* in_sizes, int n_in,
                              void* d_out, int out_size, void* d_ws, size_t ws_size,
                              hipStream_t stream) {
  const float* h      = (const float*)d_in[0];
  const int*   src    = (const int*)  d_in[1];
  const int*   dst    = (const int*)  d_in[2];
  const float* W      = (const float*)d_in[3];
  const float* attn_l = (const float*)d_in[4];
  const float* attn_r = (const float*)d_in[5];
  const float* gamma  = (const float*)d_in[6];
  const float* beta   = (const float*)d_in[7];
  float* out = (float*)d_out;

  float* ws   = (float*)d_ws;
  float* feat = ws + OFF_FEAT;
  float* el   = ws + OFF_EL;
  float* er   = ws + OFF_ER;
  float* ebuf = ws + OFF_E;
  float* emax = ws + OFF_EMAX;
  float* den  = ws + OFF_DEN;
  float* agg  = ws + OFF_AGG;
  float* mean = ws + OFF_MEAN;
  float* rstd = ws + OFF_RSTD;

  // 1) projection: 3125 blocks x 8 waves, one 16x16 WMMA tile per wave
  gemm_feat_wmma<<<3125, 256, 0, stream>>>(h, W, feat);
  // 2) attention dot products
  elr_kernel<<<(N_NODES * HEADS + 255) / 256, 256, 0, stream>>>(feat, attn_l, attn_r, el, er);
  // 3) init scratch
  init_kernel<<<(N_NODES * CH + 255) / 256, 256, 0, stream>>>(emax, den, agg);
  // 4) edge logits + segment max
  edge_logits<<<(N_EDGES * HEADS + 255) / 256, 256, 0, stream>>>(src, dst, el, er, ebuf, emax);
  // 5) exp + segment sum
  edge_exp<<<(N_EDGES * HEADS + 255) / 256, 256, 0, stream>>>(dst, ebuf, emax, den);
  // 6) weighted scatter-aggregate (102.4M threads)
  edge_aggregate<<<(int)(((long long)N_EDGES * CH + 255) / 256), 256, 0, stream>>>(
      src, dst, feat, ebuf, den, agg);
  // 7) BN stats
  bn_stats<<<CH, 256, 0, stream>>>(agg, mean, rstd);
  // 8) normalize + ELU + residual
  finalize<<<(N_NODES * CH + 255) / 256, 256, 0, stream>>>(h, agg, mean, rstd, gamma, beta, out);
}